// VoteLayer_24781961298241
// MI455X (gfx1250) — compile-verified
//
#include <hip/hip_runtime.h>
#include <math.h>

// Problem constants (from reference)
#define BATCH   8
#define CIN     256
#define NPTS    16384
#define NCLASS  3
#define NPOINT  4096
#define MIDC    128
#define EPSBN   1e-5f

typedef float v2f __attribute__((ext_vector_type(2)));
typedef float v8f __attribute__((ext_vector_type(8)));

__device__ __forceinline__ float gelu_exact(float x) {
    // gelu(x) = 0.5*x*(1+erf(x/sqrt(2)))  (approximate=False)
    return 0.5f * x * (1.0f + erff(x * 0.70710678118654752440f));
}

// ---------------------------------------------------------------------------
// Stage 1: cls_preds = W2 * gelu(bn1(W1 * F)) + b2 ; score = sigmoid(max_cls)
// One wave owns a 16-point column; 16 channel-stripes of 16; K=256 via 64
// v_wmma_f32_16x16x4_f32. h is never written to memory.
// grid = (NPTS/128, BATCH), block = 256 (8 waves)
// ---------------------------------------------------------------------------
__global__ __launch_bounds__(256)
void stage1_cls_kernel(const float* __restrict__ F,    // (B, C, N)
                       const float* __restrict__ W1,   // (C, C)
                       const float* __restrict__ g1,  const float* __restrict__ be1,
                       const float* __restrict__ mu1, const float* __restrict__ var1,
                       const float* __restrict__ W2,   // (3, C)
                       const float* __restrict__ b2,   // (3)
                       float* __restrict__ cls_out,    // (B, 3, N)
                       float* __restrict__ score)      // (B, N)
{
    __shared__ float w1s[16 * 260];            // one 16-row stripe of W1, padded pitch
    __shared__ float scale_s[CIN], shift_s[CIN];
    __shared__ float w2s[NCLASS * CIN];
    __shared__ float b2s[NCLASS];

    const int tid  = threadIdx.x;
    const int b    = blockIdx.y;
    const int lane = tid & 31;
    const int wave = tid >> 5;
    const int n    = lane & 15;   // point within 16-col tile (WMMA N)
    const int h    = lane >> 4;   // lane half (selects K pair / M+8)
    const int pt0  = blockIdx.x * 128 + wave * 16;

    // Precompute bn1 scale/shift, stage W2/b2 in LDS.
    for (int i = tid; i < CIN; i += 256) {
        float sc = g1[i] * rsqrtf(var1[i] + EPSBN);
        scale_s[i] = sc;
        shift_s[i] = be1[i] - mu1[i] * sc;
    }
    for (int i = tid; i < NCLASS * CIN; i += 256) w2s[i] = W2[i];
    if (tid < NCLASS) b2s[tid] = b2[tid];

    const float* Fb   = F + ((size_t)b * CIN) * NPTS + pt0 + n;  // column base
    const float* bcol = Fb + (size_t)(2 * h) * NPTS;             // + K half offset

    float cls0 = 0.f, cls1 = 0.f, cls2 = 0.f;

    for (int ct = 0; ct < 16; ++ct) {
        __syncthreads();
        // Cooperative load of W1 stripe rows [ct*16, ct*16+16) with pitch 260.
        for (int i = tid; i < 16 * CIN; i += 256) {
            int r = i >> 8;        // row in stripe
            int c = i & 255;       // K index
            w1s[r * 260 + c] = W1[(size_t)(ct * 16 + r) * CIN + c];
        }
        __syncthreads();

        v8f acc = {0.f, 0.f, 0.f, 0.f, 0.f, 0.f, 0.f, 0.f};
        const float* arow = &w1s[n * 260 + 2 * h];
        #pragma unroll 4
        for (int k = 0; k < 64; ++k) {
            // A: W1[ct*16 + M, 4k + 2h + {0,1}]  (16x4 f32 operand, 2 VGPRs)
            v2f a = *(const v2f*)(arow + 4 * k);
            // B: F[4k + 2h + {0,1}, pt0 + n]     (4x16 f32 operand, 2 VGPRs)
            v2f bb;
            bb.x = bcol[(size_t)(4 * k)     * NPTS];
            bb.y = bcol[(size_t)(4 * k + 1) * NPTS];
            acc = __builtin_amdgcn_wmma_f32_16x16x4_f32(
                false, a, false, bb, (short)0, acc, false, false);
        }
        // Fused epilogue: bn1 + exact gelu + W2 partial reduction (h discarded).
        #pragma unroll
        for (int r = 0; r < 8; ++r) {
            int   m  = ct * 16 + r + 8 * h;   // output channel of acc[r]
            float x  = acc[r] * scale_s[m] + shift_s[m];
            float gv = gelu_exact(x);
            cls0 += w2s[0 * CIN + m] * gv;
            cls1 += w2s[1 * CIN + m] * gv;
            cls2 += w2s[2 * CIN + m] * gv;
        }
    }
    // Combine the two lane halves (complementary channel subsets, same point).
    cls0 += __shfl_xor(cls0, 16, 32);
    cls1 += __shfl_xor(cls1, 16, 32);
    cls2 += __shfl_xor(cls2, 16, 32);

    if (h == 0) {
        int   col = pt0 + n;
        float c0 = cls0 + b2s[0];
        float c1 = cls1 + b2s[1];
        float c2 = cls2 + b2s[2];
        cls_out[((size_t)b * NCLASS + 0) * NPTS + col] = c0;
        cls_out[((size_t)b * NCLASS + 1) * NPTS + col] = c1;
        cls_out[((size_t)b * NCLASS + 2) * NPTS + col] = c2;
        float mx = fmaxf(c0, fmaxf(c1, c2));
        score[(size_t)b * NPTS + col] = 1.0f / (1.0f + expf(-mx)); // sigmoid
    }
}

// ---------------------------------------------------------------------------
// Stage 2: exact stable top-k by rank counting (matches jax.lax.top_k order:
// descending value, ties broken by ascending index). Each thread ranks one
// element against all N from LDS tiles; rank<NPOINT writes idx[rank]=i.
// grid = (NPTS/256, BATCH), block = 256
// ---------------------------------------------------------------------------
__global__ __launch_bounds__(256)
void topk_rank_kernel(const float* __restrict__ score,  // (B, N)
                      int* __restrict__ idx_out)        // (B, NPOINT)
{
    __shared__ float sb[2048];
    const int   b  = blockIdx.y;
    const int   i  = blockIdx.x * 256 + threadIdx.x;
    const float si = score[(size_t)b * NPTS + i];

    int rank = 0;
    for (int j0 = 0; j0 < NPTS; j0 += 2048) {
        __syncthreads();
        for (int t = threadIdx.x; t < 2048; t += 256)
            sb[t] = score[(size_t)b * NPTS + j0 + t];
        __syncthreads();
        #pragma unroll 8
        for (int t = 0; t < 2048; ++t) {
            float sj = sb[t];
            int   j  = j0 + t;
            rank += (int)((sj > si) || ((sj == si) && (j < i)));
        }
    }
    if (rank < NPOINT) idx_out[(size_t)b * NPOINT + rank] = i;
}

// ---------------------------------------------------------------------------
// Stage 3: gather + r = gelu(bn2(W3 * G)) ; offsets = W4*r + b4 ; clamp ;
// preds = origins + limited. Gather folded into B-operand addressing.
// grid = (NPOINT/128, BATCH), block = 256 (8 waves)
// ---------------------------------------------------------------------------
__global__ __launch_bounds__(256)
void stage2_vote_kernel(const float* __restrict__ F,      // (B, C, N)
                        const float* __restrict__ points, // (B, N, 3)
                        const int*   __restrict__ idx,    // (B, NPOINT)
                        const float* __restrict__ W3,     // (MID, C)
                        const float* __restrict__ g2,  const float* __restrict__ be2,
                        const float* __restrict__ mu2, const float* __restrict__ var2,
                        const float* __restrict__ W4,     // (3, MID)
                        const float* __restrict__ b4,     // (3)
                        const float* __restrict__ lim,    // (3)
                        float* __restrict__ ctr_preds,    // (B, NPOINT, 3)
                        float* __restrict__ ctr_origins,  // (B, NPOINT, 3)
                        float* __restrict__ ctr_offsets)  // (B, NPOINT, 3)
{
    __shared__ float w3s[16 * 260];
    __shared__ float scale_s[MIDC], shift_s[MIDC];
    __shared__ float w4s[3 * MIDC];
    __shared__ float b4s[3], lims[3];

    const int tid  = threadIdx.x;
    const int b    = blockIdx.y;
    const int lane = tid & 31;
    const int wave = tid >> 5;
    const int n    = lane & 15;
    const int h    = lane >> 4;
    const int pt0  = blockIdx.x * 128 + wave * 16;
    const int j    = pt0 + n;                 // output point slot

    for (int i = tid; i < MIDC; i += 256) {
        float sc = g2[i] * rsqrtf(var2[i] + EPSBN);
        scale_s[i] = sc;
        shift_s[i] = be2[i] - mu2[i] * sc;
    }
    for (int i = tid; i < 3 * MIDC; i += 256) w4s[i] = W4[i];
    if (tid < 3) { b4s[tid] = b4[tid]; lims[tid] = lim[tid]; }

    const int    src  = idx[(size_t)b * NPOINT + j];          // gathered source point
    const float* Gb   = F + ((size_t)b * CIN) * NPTS + src;
    const float* bcol = Gb + (size_t)(2 * h) * NPTS;

    float off0 = 0.f, off1 = 0.f, off2 = 0.f;
    for (int ct = 0; ct < 8; ++ct) {
        __syncthreads();
        for (int i = tid; i < 16 * CIN; i += 256) {
            int r = i >> 8, c = i & 255;
            w3s[r * 260 + c] = W3[(size_t)(ct * 16 + r) * CIN + c];
        }
        __syncthreads();

        v8f acc = {0.f, 0.f, 0.f, 0.f, 0.f, 0.f, 0.f, 0.f};
        const float* arow = &w3s[n * 260 + 2 * h];
        #pragma unroll 4
        for (int k = 0; k < 64; ++k) {
            v2f a = *(const v2f*)(arow + 4 * k);
            v2f bb;
            bb.x = bcol[(size_t)(4 * k)     * NPTS];
            bb.y = bcol[(size_t)(4 * k + 1) * NPTS];
            acc = __builtin_amdgcn_wmma_f32_16x16x4_f32(
                false, a, false, bb, (short)0, acc, false, false);
        }
        #pragma unroll
        for (int r = 0; r < 8; ++r) {
            int   m  = ct * 16 + r + 8 * h;
            float x  = acc[r] * scale_s[m] + shift_s[m];
            float gv = gelu_exact(x);
            off0 += w4s[0 * MIDC + m] * gv;
            off1 += w4s[1 * MIDC + m] * gv;
            off2 += w4s[2 * MIDC + m] * gv;
        }
    }
    off0 += __shfl_xor(off0, 16, 32);
    off1 += __shfl_xor(off1, 16, 32);
    off2 += __shfl_xor(off2, 16, 32);

    if (h == 0) {
        float off[3] = { off0 + b4s[0], off1 + b4s[1], off2 + b4s[2] };
        size_t obase = ((size_t)b * NPOINT + j) * 3;
        #pragma unroll
        for (int d = 0; d < 3; ++d) {
            float o  = points[((size_t)b * NPTS + src) * 3 + d];
            float L  = lims[d];
            float lm = fminf(off[d], L);
            lm       = fmaxf(lm, -L);
            ctr_offsets[obase + d] = off[d];
            ctr_origins[obase + d] = o;
            ctr_preds  [obase + d] = o + lm;
        }
    }
}

// ---------------------------------------------------------------------------
extern "C" void kernel_launch(void* const* d_in, const int* in_sizes, int n_in,
                              void* d_out, int out_size, void* d_ws, size_t ws_size,
                              hipStream_t stream) {
    (void)in_sizes; (void)n_in; (void)out_size; (void)ws_size;

    const float* points = (const float*)d_in[0];   // (B, N, 3)
    const float* feats  = (const float*)d_in[1];   // (B, C, N)
    const float* W1     = (const float*)d_in[2];
    const float* g1     = (const float*)d_in[3];
    const float* be1    = (const float*)d_in[4];
    const float* mu1    = (const float*)d_in[5];
    const float* var1   = (const float*)d_in[6];
    const float* W2     = (const float*)d_in[7];
    const float* b2     = (const float*)d_in[8];
    const float* W3     = (const float*)d_in[9];
    const float* g2     = (const float*)d_in[10];
    const float* be2    = (const float*)d_in[11];
    const float* mu2    = (const float*)d_in[12];
    const float* var2   = (const float*)d_in[13];
    const float* W4     = (const float*)d_in[14];
    const float* b4     = (const float*)d_in[15];
    const float* lim    = (const float*)d_in[16];

    float* out = (float*)d_out;
    // Output layout (flat, return order):
    float* cls_preds   = out;                                   // 8*3*16384
    float* ctr_preds   = out + (size_t)BATCH * NCLASS * NPTS;   // +393216
    float* ctr_origins = ctr_preds + (size_t)BATCH * NPOINT * 3;// +98304
    float* ctr_offsets = ctr_origins + (size_t)BATCH * NPOINT * 3;

    // Workspace: score (B*N f32) then idx (B*NPOINT i32)
    float* score = (float*)d_ws;
    int*   idxws = (int*)((char*)d_ws + (size_t)BATCH * NPTS * sizeof(float));

    stage1_cls_kernel<<<dim3(NPTS / 128, BATCH), 256, 0, stream>>>(
        feats, W1, g1, be1, mu1, var1, W2, b2, cls_preds, score);

    topk_rank_kernel<<<dim3(NPTS / 256, BATCH), 256, 0, stream>>>(score, idxws);

    stage2_vote_kernel<<<dim3(NPOINT / 128, BATCH), 256, 0, stream>>>(
        feats, points, idxws, W3, g2, be2, mu2, var2, W4, b4, lim,
        ctr_preds, ctr_origins, ctr_offsets);
}